// Scheduler_88562225644054
// MI455X (gfx1250) — compile-verified
//
#include <hip/hip_runtime.h>
#include <math.h>

// ---------------------------------------------------------------------------
// Problem constants (from the reference)
// ---------------------------------------------------------------------------
#define NS    4096      // N_SRC
#define NT    4096      // N_TGT
#define DF    256       // D_FEAT
#define HID   64
#define OUTC  32
#define NTOT  (NS + NT) // 8192
// rank indices for quantile(0.9) with linear interpolation over 16,777,216 vals
#define RANK1 15099493u
#define RANK2 15099494u

typedef float v2f __attribute__((ext_vector_type(2)));
typedef float v8f __attribute__((ext_vector_type(8)));

// fp32 WMMA: D(16x16) += A(16x4) * B(4x16)
__device__ __forceinline__ v8f wmma_step(v2f a, v2f b, v8f c) {
  return __builtin_amdgcn_wmma_f32_16x16x4_f32(
      /*neg_a=*/false, a, /*neg_b=*/false, b,
      /*c_mod=*/(short)0, c, /*reuse_a=*/false, /*reuse_b=*/false);
}

// ---------------------------------------------------------------------------
// 1) scores = relu(S @ T^T)   (one 16x16 tile per wave, K-loop over 256)
//    B = T^T, so B[k][n] = T[n][k]: B-fragment loads are contiguous rows of T.
// ---------------------------------------------------------------------------
__global__ void scores_kernel(const float* __restrict__ S,
                              const float* __restrict__ T,
                              float* __restrict__ scores) {
  const int lane = threadIdx.x;            // blockDim.x == 32
  const int m0 = blockIdx.x * 16;
  const int n0 = blockIdx.y * 16;
  const int r  = lane & 15;
  const int kk = (lane >> 4) * 2;          // K sub-offset per lane half
  const float* arow = S + (size_t)(m0 + r) * DF;
  const float* brow = T + (size_t)(n0 + r) * DF;
  v8f acc = {};
  for (int k = 0; k < DF; k += 4) {
    v2f a = *(const v2f*)(arow + k + kk);
    v2f b = *(const v2f*)(brow + k + kk);
    acc = wmma_step(a, b, acc);
  }
  const int mh = (lane >> 4) * 8;
#pragma unroll
  for (int i = 0; i < 8; ++i) {
    float v = acc[i];
    v = v > 0.0f ? v : 0.0f;               // relu
    scores[(size_t)(m0 + mh + i) * NT + (n0 + (lane & 15))] = v;
  }
}

// ---------------------------------------------------------------------------
// 2) Exact quantile via 2-level radix select on u32 bit patterns (all >= 0)
// ---------------------------------------------------------------------------
__global__ void zero_u32_kernel(unsigned* p, int n) {
  int i = blockIdx.x * blockDim.x + threadIdx.x;
  if (i < n) p[i] = 0u;
}

__global__ void histo_hi_kernel(const float* __restrict__ scores,
                                unsigned* __restrict__ hist) {
  const size_t n = (size_t)NS * NT;
  for (size_t i = blockIdx.x * (size_t)blockDim.x + threadIdx.x; i < n;
       i += (size_t)gridDim.x * blockDim.x) {
    unsigned bits = __float_as_uint(scores[i]);
    atomicAdd(&hist[bits >> 16], 1u);
  }
}

__global__ void scan_hi_kernel(const unsigned* __restrict__ hist,
                               unsigned rank, unsigned* __restrict__ sel) {
  unsigned cum = 0;
  for (int b = 0; b < 65536; ++b) {
    unsigned c = hist[b];
    if (cum + c > rank) { sel[0] = (unsigned)b; sel[1] = cum; return; }
    cum += c;
  }
  sel[0] = 65535u; sel[1] = cum;
}

__global__ void histo_lo_kernel(const float* __restrict__ scores,
                                const unsigned* __restrict__ sel,
                                unsigned* __restrict__ hist) {
  const unsigned hibin = sel[0];
  const size_t n = (size_t)NS * NT;
  for (size_t i = blockIdx.x * (size_t)blockDim.x + threadIdx.x; i < n;
       i += (size_t)gridDim.x * blockDim.x) {
    unsigned bits = __float_as_uint(scores[i]);
    if ((bits >> 16) == hibin) atomicAdd(&hist[bits & 0xFFFFu], 1u);
  }
}

__global__ void scan_lo_kernel(const unsigned* __restrict__ hist,
                               unsigned rank, const unsigned* __restrict__ sel,
                               float* __restrict__ v_out) {
  unsigned cum = sel[1];
  const unsigned hibin = sel[0];
  for (int b = 0; b < 65536; ++b) {
    unsigned c = hist[b];
    if (cum + c > rank) {
      *v_out = __uint_as_float((hibin << 16) | (unsigned)b);
      return;
    }
    cum += c;
  }
  *v_out = __uint_as_float((hibin << 16) | 0xFFFFu);
}

__global__ void combine_md_kernel(const float* v1, const float* v2, float* md) {
  *md = 0.5f * (*v1 + *v2);
}

// ---------------------------------------------------------------------------
// 3) Degrees and dinv
// ---------------------------------------------------------------------------
__global__ void deg_row_kernel(const float* __restrict__ scores,
                               const float* __restrict__ md_p,
                               unsigned* __restrict__ deg) {
  __shared__ unsigned sm[256];
  const int row = blockIdx.x;
  const float md = *md_p;
  unsigned c = 0;
  for (int j = threadIdx.x; j < NT; j += 256)
    c += (scores[(size_t)row * NT + j] > md) ? 1u : 0u;
  sm[threadIdx.x] = c;
  __syncthreads();
  for (int o = 128; o > 0; o >>= 1) {
    if ((int)threadIdx.x < o) sm[threadIdx.x] += sm[threadIdx.x + o];
    __syncthreads();
  }
  if (threadIdx.x == 0) deg[row] = sm[0];
}

__global__ void deg_col_kernel(const float* __restrict__ scores,
                               const float* __restrict__ md_p,
                               unsigned* __restrict__ deg) {
  const int j = blockIdx.x * blockDim.x + threadIdx.x;  // 0..NT-1
  if (j >= NT) return;
  const float md = *md_p;
  unsigned c = 0;
  for (int i = 0; i < NS; ++i)
    c += (scores[(size_t)i * NT + j] > md) ? 1u : 0u;
  deg[j] = c;
}

__global__ void dinv_kernel(const unsigned* __restrict__ deg_src,
                            const unsigned* __restrict__ deg_tgt,
                            float* __restrict__ dinv) {
  int i = blockIdx.x * blockDim.x + threadIdx.x;
  if (i >= NTOT) return;
  unsigned d = (i < NS) ? deg_src[i] : deg_tgt[i - NS];
  dinv[i] = rsqrtf((float)d + 1.0f);       // +1 for the self loop
}

// ---------------------------------------------------------------------------
// 4) y1 = concat(S,T) @ W1   (fp32 WMMA, K=256, N=64)
// ---------------------------------------------------------------------------
__global__ void gemm_x_w1_kernel(const float* __restrict__ S,
                                 const float* __restrict__ T,
                                 const float* __restrict__ W,
                                 float* __restrict__ Y) {
  const int lane = threadIdx.x;
  const int m0 = blockIdx.x * 16;
  const int n0 = blockIdx.y * 16;
  const int r = lane & 15, kk = (lane >> 4) * 2, c = lane & 15;
  const int m = m0 + r;
  const float* arow = (m < NS) ? (S + (size_t)m * DF) : (T + (size_t)(m - NS) * DF);
  v8f acc = {};
  for (int k = 0; k < DF; k += 4) {
    v2f a = *(const v2f*)(arow + k + kk);
    v2f b;
    b.x = W[(size_t)(k + kk) * HID + n0 + c];
    b.y = W[(size_t)(k + kk + 1) * HID + n0 + c];
    acc = wmma_step(a, b, acc);
  }
  const int mh = (lane >> 4) * 8;
#pragma unroll
  for (int i = 0; i < 8; ++i)
    Y[(size_t)(m0 + mh + i) * HID + n0 + c] = acc[i];
}

// Generic NN GEMM (row-major A[MxK], B[KxN]) — used for y2 = h @ W2
__global__ void gemm_nn_kernel(const float* __restrict__ A, int lda,
                               const float* __restrict__ B, int ldb,
                               float* __restrict__ C, int ldc, int K) {
  const int lane = threadIdx.x;
  const int m0 = blockIdx.x * 16;
  const int n0 = blockIdx.y * 16;
  const int r = lane & 15, kk = (lane >> 4) * 2, c = lane & 15;
  const float* arow = A + (size_t)(m0 + r) * lda;
  v8f acc = {};
  for (int k = 0; k < K; k += 4) {
    v2f a = *(const v2f*)(arow + k + kk);
    v2f b;
    b.x = B[(size_t)(k + kk) * ldb + n0 + c];
    b.y = B[(size_t)(k + kk + 1) * ldb + n0 + c];
    acc = wmma_step(a, b, acc);
  }
  const int mh = (lane >> 4) * 8;
#pragma unroll
  for (int i = 0; i < 8; ++i)
    C[(size_t)(m0 + mh + i) * ldc + n0 + c] = acc[i];
}

// ---------------------------------------------------------------------------
// 5) z = dinv ⊙ y  (elementwise)
// ---------------------------------------------------------------------------
__global__ void zmul_kernel(const float* __restrict__ y,
                            const float* __restrict__ dinv,
                            float* __restrict__ z, int ncols, int ntotal) {
  int idx = blockIdx.x * blockDim.x + threadIdx.x;
  if (idx < ntotal) z[idx] = dinv[idx / ncols] * y[idx];
}

// ---------------------------------------------------------------------------
// 6) GCN aggregation: out[row] = dinv[row]*(z[row] + Mask @ z_other) + bias
//    Mask A-fragment built ONCE per k-step from scores > md and reused for all
//    NCOLS/16 column tiles (multi-accumulator): cuts scores (L2) traffic by
//    NCOLS/16x and raises WMMA density in the hot loop. Prefetch the next
//    scores cache lines (global_prefetch_b8) — the mask stream is predictable.
// ---------------------------------------------------------------------------
template <int NCOLS, int DO_RELU>
__global__ void agg_kernel_t(const float* __restrict__ scores,
                             const float* __restrict__ md_p,
                             const float* __restrict__ z,
                             const float* __restrict__ dinv,
                             const float* __restrict__ bias,
                             float* __restrict__ out) {
  constexpr int NTILES = NCOLS / 16;
  const int lane = threadIdx.x;
  const int row0 = blockIdx.x * 16;        // 0..8191 step 16 (block-uniform)
  const float md = *md_p;
  const int r = lane & 15, kk = (lane >> 4) * 2, c = lane & 15;
  v8f acc[NTILES];
#pragma unroll
  for (int t = 0; t < NTILES; ++t) acc[t] = (v8f){};

  if (row0 < NS) {                         // source rows: Mask @ z_t
    const float* srow = scores + (size_t)(row0 + r) * NT;
    const float* zt = z + (size_t)NS * NCOLS;
    for (int k = 0; k < NT; k += 4) {
      if ((k & 63) == 0)                   // prefetch next 256B chunk of mask row
        __builtin_prefetch(srow + k + 64, 0, 1);
      v2f sv = *(const v2f*)(srow + k + kk);
      v2f a;
      a.x = (sv.x > md) ? 1.0f : 0.0f;
      a.y = (sv.y > md) ? 1.0f : 0.0f;
#pragma unroll
      for (int t = 0; t < NTILES; ++t) {
        v2f b;
        b.x = zt[(size_t)(k + kk) * NCOLS + t * 16 + c];
        b.y = zt[(size_t)(k + kk + 1) * NCOLS + t * 16 + c];
        acc[t] = wmma_step(a, b, acc[t]);
      }
    }
  } else {                                 // target rows: Mask^T @ z_s
    const int j = (row0 - NS) + r;
    for (int k = 0; k < NS; k += 4) {
      v2f a;
      a.x = (scores[(size_t)(k + kk) * NT + j] > md) ? 1.0f : 0.0f;
      a.y = (scores[(size_t)(k + kk + 1) * NT + j] > md) ? 1.0f : 0.0f;
#pragma unroll
      for (int t = 0; t < NTILES; ++t) {
        v2f b;
        b.x = z[(size_t)(k + kk) * NCOLS + t * 16 + c];
        b.y = z[(size_t)(k + kk + 1) * NCOLS + t * 16 + c];
        acc[t] = wmma_step(a, b, acc[t]);
      }
    }
  }

  const int mh = (lane >> 4) * 8;
#pragma unroll
  for (int t = 0; t < NTILES; ++t) {
#pragma unroll
    for (int i = 0; i < 8; ++i) {
      const int row = row0 + mh + i;
      const int col = t * 16 + c;
      float v = dinv[row] * (z[(size_t)row * NCOLS + col] + acc[t][i]) + bias[col];
      if (DO_RELU) v = v > 0.0f ? v : 0.0f;
      out[(size_t)row * NCOLS + col] = v;
    }
  }
}

// ---------------------------------------------------------------------------
// 7) target = mean(emb[NS:], axis=0)   (32 columns, one thread each)
// ---------------------------------------------------------------------------
__global__ void tmean_kernel(const float* __restrict__ emb,
                             float* __restrict__ target) {
  const int c = threadIdx.x;               // blockDim.x == 32
  float s = 0.0f;
  for (int rr = 0; rr < NT; ++rr)
    s += emb[(size_t)(NS + rr) * OUTC + c];
  target[c] = s * (1.0f / (float)NT);
}

// ---------------------------------------------------------------------------
// 8) head: out[i] = 0.5*sigmoid((emb_src[i] ⊙ target) @ w + b) + 0.5*sigmoid(tv)
// ---------------------------------------------------------------------------
__global__ void final_kernel(const float* __restrict__ emb,
                             const float* __restrict__ target,
                             const float* __restrict__ w,
                             const float* __restrict__ bptr,
                             const float* __restrict__ task_vec,
                             float* __restrict__ out) {
  const int i = blockIdx.x * blockDim.x + threadIdx.x;
  if (i >= NS) return;
  float s = 0.0f;
#pragma unroll
  for (int c = 0; c < OUTC; ++c)
    s += emb[(size_t)i * OUTC + c] * target[c] * w[c];
  s += bptr[0];
  float sg1 = 1.0f / (1.0f + expf(-s));
  float sg2 = 1.0f / (1.0f + expf(-task_vec[i]));
  out[i] = 0.5f * sg1 + 0.5f * sg2;
}

// ---------------------------------------------------------------------------
// Launch
// ---------------------------------------------------------------------------
extern "C" void kernel_launch(void* const* d_in, const int* in_sizes, int n_in,
                              void* d_out, int out_size, void* d_ws, size_t ws_size,
                              hipStream_t stream) {
  (void)in_sizes; (void)n_in; (void)out_size; (void)ws_size;
  const float* S   = (const float*)d_in[0];
  const float* T   = (const float*)d_in[1];
  const float* W1  = (const float*)d_in[2];
  const float* b1  = (const float*)d_in[3];
  const float* W2  = (const float*)d_in[4];
  const float* b2  = (const float*)d_in[5];
  const float* w   = (const float*)d_in[6];
  const float* b   = (const float*)d_in[7];
  const float* tv  = (const float*)d_in[8];
  float* out = (float*)d_out;

  // Workspace layout (assumes ws_size >= ~74 MB)
  char* ws = (char*)d_ws;
  float* scores = (float*)ws;                                   // 4096*4096 f32
  float* y1  = (float*)(ws + (size_t)NS * NT * sizeof(float));  // 8192*64
  float* z1  = y1 + (size_t)NTOT * HID;                         // 8192*64
  float* h   = z1 + (size_t)NTOT * HID;                         // 8192*64
  float* y2  = h + (size_t)NTOT * HID;                          // 8192*32
  float* z2  = y2 + (size_t)NTOT * OUTC;                        // 8192*32
  float* emb = z2 + (size_t)NTOT * OUTC;                        // 8192*32
  unsigned* deg_src = (unsigned*)(emb + (size_t)NTOT * OUTC);   // 4096
  unsigned* deg_tgt = deg_src + NS;                             // 4096
  float* dinv = (float*)(deg_tgt + NT);                         // 8192
  unsigned* hist = (unsigned*)(dinv + NTOT);                    // 65536
  unsigned* sel  = hist + 65536;                                // 4
  float* v1  = (float*)(sel + 4);
  float* v2  = v1 + 1;
  float* md  = v2 + 1;
  float* tgt = md + 1;                                          // 32

  const dim3 wave(32);

  // 1) scores = relu(S @ T^T)
  scores_kernel<<<dim3(NS / 16, NT / 16), wave, 0, stream>>>(S, T, scores);

  // 2) exact quantile(0.9): radix select for ranks RANK1, RANK2
  zero_u32_kernel<<<256, 256, 0, stream>>>(hist, 65536);
  histo_hi_kernel<<<2048, 256, 0, stream>>>(scores, hist);
  scan_hi_kernel<<<1, 1, 0, stream>>>(hist, RANK1, sel);
  zero_u32_kernel<<<256, 256, 0, stream>>>(hist, 65536);
  histo_lo_kernel<<<2048, 256, 0, stream>>>(scores, sel, hist);
  scan_lo_kernel<<<1, 1, 0, stream>>>(hist, RANK1, sel, v1);

  zero_u32_kernel<<<256, 256, 0, stream>>>(hist, 65536);
  histo_hi_kernel<<<2048, 256, 0, stream>>>(scores, hist);
  scan_hi_kernel<<<1, 1, 0, stream>>>(hist, RANK2, sel);
  zero_u32_kernel<<<256, 256, 0, stream>>>(hist, 65536);
  histo_lo_kernel<<<2048, 256, 0, stream>>>(scores, sel, hist);
  scan_lo_kernel<<<1, 1, 0, stream>>>(hist, RANK2, sel, v2);

  combine_md_kernel<<<1, 1, 0, stream>>>(v1, v2, md);

  // 3) degrees + dinv
  deg_row_kernel<<<NS, 256, 0, stream>>>(scores, md, deg_src);
  deg_col_kernel<<<NT / 256, 256, 0, stream>>>(scores, md, deg_tgt);
  dinv_kernel<<<NTOT / 256, 256, 0, stream>>>(deg_src, deg_tgt, dinv);

  // 4) layer 1: y1 = x @ W1 ; z1 = dinv*y1 ; h = relu(dinv*(z1 + M z1_other) + b1)
  gemm_x_w1_kernel<<<dim3(NTOT / 16, HID / 16), wave, 0, stream>>>(S, T, W1, y1);
  zmul_kernel<<<(NTOT * HID + 255) / 256, 256, 0, stream>>>(y1, dinv, z1, HID, NTOT * HID);
  agg_kernel_t<HID, 1><<<dim3(NTOT / 16), wave, 0, stream>>>(scores, md, z1, dinv, b1, h);

  // 5) layer 2: y2 = h @ W2 ; z2 = dinv*y2 ; emb = dinv*(z2 + M z2_other) + b2
  gemm_nn_kernel<<<dim3(NTOT / 16, OUTC / 16), wave, 0, stream>>>(h, HID, W2, OUTC, y2, OUTC, HID);
  zmul_kernel<<<(NTOT * OUTC + 255) / 256, 256, 0, stream>>>(y2, dinv, z2, OUTC, NTOT * OUTC);
  agg_kernel_t<OUTC, 0><<<dim3(NTOT / 16), wave, 0, stream>>>(scores, md, z2, dinv, b2, emb);

  // 6) head
  tmean_kernel<<<1, 32, 0, stream>>>(emb, tgt);
  final_kernel<<<NS / 256, 256, 0, stream>>>(emb, tgt, w, b, tv, out);
}